// DeformUnfold3D_37855841747687
// MI455X (gfx1250) — compile-verified
//
#include <hip/hip_runtime.h>

// Problem constants (from reference): kernel 3x3x3, stride 1, pad 1, dil 1
#define B_N    2
#define C_CH   16
#define D_DIM  24
#define H_DIM  48
#define W_DIM  48
#define KTAPS  27
#define P_TOT  (D_DIM * H_DIM * W_DIM)   // 55296; Do==D etc.
#define HALF_P (P_TOT / 2)               // 27648
#define DHW    P_TOT
#define KP     (KTAPS * P_TOT)

typedef float f2g __attribute__((ext_vector_type(2), aligned(4))); // 4B-aligned gather pair
typedef float f2  __attribute__((ext_vector_type(2)));             // naturally 8B-aligned pair

__global__ __launch_bounds__(256) void DeformUnfold3D_kernel(
    const float* __restrict__ x,     // [B, C, D, H, W]
    const float* __restrict__ off,   // [B, K, 3, P]
    float* __restrict__ out,         // [B, C*K, P]
    int totalThreads)
{
    unsigned tid = blockIdx.x * 256u + threadIdx.x;   // one thread = 2 consecutive p
    if (tid >= (unsigned)totalThreads) return;

    unsigned pp = tid % (unsigned)HALF_P;
    unsigned bk = tid / (unsigned)HALF_P;             // b*K + k
    unsigned k  = bk % (unsigned)KTAPS;
    unsigned b  = bk / (unsigned)KTAPS;
    unsigned p  = pp * 2u;                            // even; p and p+1 share od/oh row

    unsigned ow = p % (unsigned)W_DIM;                // even <= 46, so ow+1 stays in-row
    unsigned t1 = p / (unsigned)W_DIM;
    unsigned oh = t1 % (unsigned)H_DIM;
    unsigned od = t1 / (unsigned)H_DIM;

    unsigned kk = k % 3u;
    unsigned kj = (k / 3u) % 3u;
    unsigned ki = k / 9u;

    // streamed read-once offsets: paired non-temporal b64 loads
    const float* offptr = off + (size_t)bk * 3u * P_TOT + p;
    f2 offd = __builtin_nontemporal_load((const f2*)(offptr));
    f2 offh = __builtin_nontemporal_load((const f2*)(offptr + P_TOT));
    f2 offw = __builtin_nontemporal_load((const f2*)(offptr + 2 * P_TOT));

    float bd = (float)((int)od + (int)ki - 1);
    float bh = (float)((int)oh + (int)kj - 1);
    float bw = (float)((int)ow + (int)kk - 1);

    // Per sub-position u (p+u): 4 (d,h) row-pairs, each gathered as one b64 at
    // {wb, wb+1}; W-border handling folded into pre-computed weights (a, bwt).
    int   pbase[2][4];
    float pa[2][4], pbw[2][4];

    #pragma unroll
    for (int u = 0; u < 2; ++u) {
        float pd = bd + offd[u];
        float ph = bh + offh[u];
        float pw = (bw + (float)u) + offw[u];

        float d0f = floorf(pd); float fd = pd - d0f; int d0 = (int)d0f;
        float h0f = floorf(ph); float fh = ph - h0f; int h0 = (int)h0f;
        float w0f = floorf(pw); float fw = pw - w0f; int w0 = (int)w0f;

        bool  v0w = (w0 >= 0) && (w0 < W_DIM);        // corner w0 valid
        bool  v1w = (w0 >= -1) && (w0 < W_DIM - 1);   // corner w0+1 valid
        int   wb  = min(max(w0, 0), W_DIM - 2);       // pair base (always safe b64)
        float fw0 = 1.0f - fw;

        #pragma unroll
        for (int ci = 0; ci < 2; ++ci) {
            int   ds = d0 + ci;
            float wd = ci ? fd : (1.0f - fd);
            bool  vd = (ds >= 0) && (ds < D_DIM);
            int   dc = min(max(ds, 0), D_DIM - 1);
            #pragma unroll
            for (int cj = 0; cj < 2; ++cj) {
                int   hs = h0 + cj;
                float wh = cj ? fh : (1.0f - fh);
                bool  vh = (hs >= 0) && (hs < H_DIM);
                int   hc = min(max(hs, 0), H_DIM - 1);
                int   j  = ci * 2 + cj;

                float g   = (vd && vh) ? (wd * wh) : 0.0f;
                float wt0 = v0w ? (fw0 * g) : 0.0f;
                float wt1 = v1w ? (fw * g) : 0.0f;
                // Border remap: when w0<0 the needed (only nonzero-wt) value sits in
                // the low slot; when w0>=W-1 it sits in the high slot. The displaced
                // term always has weight 0, so results match the reference exactly.
                pa[u][j]  = (w0 < 0) ? wt1  : ((w0 >= W_DIM - 1) ? 0.0f : wt0);
                pbw[u][j] = (w0 < 0) ? 0.0f : ((w0 >= W_DIM - 1) ? wt0  : wt1);
                pbase[u][j] = (dc * H_DIM + hc) * W_DIM + wb;
            }
        }
    }

    // x (7.1 MB total) stays hot in L2; gathers use default (RT) cache policy.
    const float* xb = x + (size_t)b * C_CH * DHW;
    float*       ob = out + ((size_t)(b * C_CH) * KTAPS + k) * P_TOT + p;

    #pragma unroll 2
    for (int c = 0; c < C_CH; ++c) {
        const float* xc = xb + (size_t)c * DHW;
        f2 res;
        #pragma unroll
        for (int u = 0; u < 2; ++u) {
            f2g v0 = *(const f2g*)(xc + pbase[u][0]);
            f2g v1 = *(const f2g*)(xc + pbase[u][1]);
            f2g v2 = *(const f2g*)(xc + pbase[u][2]);
            f2g v3 = *(const f2g*)(xc + pbase[u][3]);
            float acc;
            acc  = pa[u][0] * v0.x;   acc += pbw[u][0] * v0.y;   // (d0,h0,w0),(d0,h0,w1)
            acc += pa[u][1] * v1.x;   acc += pbw[u][1] * v1.y;   // (d0,h1,*)
            acc += pa[u][2] * v2.x;   acc += pbw[u][2] * v2.y;   // (d1,h0,*)
            acc += pa[u][3] * v3.x;   acc += pbw[u][3] * v3.y;   // (d1,h1,*)
            res[u] = acc;
        }
        // streamed write-once output: paired non-temporal b64 store
        __builtin_nontemporal_store(res, (f2*)(ob + (size_t)c * KP));
    }
}

extern "C" void kernel_launch(void* const* d_in, const int* in_sizes, int n_in,
                              void* d_out, int out_size, void* d_ws, size_t ws_size,
                              hipStream_t stream) {
    const float* x   = (const float*)d_in[0];   // [2,16,24,48,48] fp32
    const float* off = (const float*)d_in[1];   // [2,81,24,48,48] fp32
    float*       out = (float*)d_out;           // [2,432,55296] fp32

    const int totalThreads = B_N * KTAPS * HALF_P;   // 1,492,992
    const int block = 256;                           // 8 x wave32
    const int grid  = (totalThreads + block - 1) / block;  // 5832, exact
    DeformUnfold3D_kernel<<<grid, block, 0, stream>>>(x, off, out, totalThreads);
}